// JacobiPool_7825430413483
// MI455X (gfx1250) — compile-verified
//
#include <hip/hip_runtime.h>
#include <hip/hip_bf16.h>
#include <hip/hip_fp16.h>
#include <math.h>

#define TPB 256
#define NEG_SLOPE 0.2f
#define EPSF 1e-12f
#define ROWPITCH 260   // LDS dwords per staged row: 256 + 4 pad (bank de-conflict)

typedef __attribute__((ext_vector_type(2)))  float    v2f;
typedef __attribute__((ext_vector_type(8)))  float    v8f;
typedef __attribute__((ext_vector_type(4)))  unsigned v4u;
typedef __attribute__((ext_vector_type(8)))  unsigned v8u;
typedef unsigned long long ull;

// ---------- order-preserving float<->uint flip (for atomic segment-max / sort keys)
__device__ __forceinline__ unsigned fflip(float f) {
  unsigned u = __float_as_uint(f);
  return (u & 0x80000000u) ? ~u : (u | 0x80000000u);
}
__device__ __forceinline__ float funflip(unsigned u) {
  u = (u & 0x80000000u) ? (u & 0x7FFFFFFFu) : ~u;
  return __uint_as_float(u);
}

// ---------- fused dual GEMV via TDM + WMMA -------------------------------------
// Computes x@W_att -> (a_s,a_d) and x@lin_W+b -> x_t in ONE pass over x.
// Each wave owns a 16-row tile of x:
//   1) TDM (tensor_load_to_lds) DMAs the 16x256 f32 tile into LDS, padded +4
//      dwords/row so the 16 A rows land in 16 distinct LDS banks.
//   2) K is swept 4 at a time with V_WMMA_F32_16X16X4_F32; B is a Kx16 slab with
//      only columns 0 (W_att) and 1 (lin_W) nonzero.
__global__ __launch_bounds__(256) void jp_dual_gemv_wmma(
    const float* __restrict__ x, const float* __restrict__ W_att,
    const float* __restrict__ lin_W, const float* __restrict__ att_src,
    const float* __restrict__ att_dst, const float* __restrict__ lin_b,
    float* __restrict__ a_s, float* __restrict__ a_d, float* __restrict__ x_t,
    int n) {
  extern __shared__ float lds_tile[];   // 8 waves * 16 rows * ROWPITCH floats
  const int lane = threadIdx.x & 31;
  const int wave = threadIdx.x >> 5;
  const int half = lane >> 4;   // 0: lanes 0-15 (K 0,1), 1: lanes 16-31 (K 2,3)
  const int l15  = lane & 15;
  long rowBase = (long)blockIdx.x * 128 + (long)wave * 16;
  if (rowBase > (long)n - 16) rowBase = (long)n - 16;   // clamp tail (dup writes ok)

  v8f acc = {};
#if defined(__gfx1250__) && __has_builtin(__builtin_amdgcn_wmma_f32_16x16x4_f32)
  {
    // ---- Tensor DMA descriptor (D#), 2D: 16x256 f32 tile of x into LDS.
    const unsigned ldsBase = (unsigned)wave * (16u * ROWPITCH * 4u);
    ull ga = (ull)(const float*)(x + rowBase * 256);
    unsigned w0 = 1u;                                            // g0: count=1 (user D#)
    unsigned w1 = ldsBase;                                       // g0: lds_addr
    unsigned w2 = (unsigned)ga;                                  // g0: global_addr lo
    unsigned w3 = (unsigned)((ga >> 32) & 0x01FFFFFFull) | (2u << 30); // addr hi | type=2
    unsigned u0 = (2u << 16) | (1u << 20) | (7u << 22) | (3u << 25);
    // g1: wg_mask=0 | data_size=4B | pad_enable | pad_interval=256dw | pad_amount=4dw
    unsigned u1 = (256u & 0xFFFFu) << 16;                        // tensor_dim0 lo
    unsigned u2 = (256u >> 16) | (((unsigned)n & 0xFFFFu) << 16);// td0 hi | tensor_dim1 lo
    unsigned u3 = (((unsigned)n >> 16) & 0xFFFFu) | (256u << 16);// td1 hi | tile_dim0=256
    unsigned u4 = 16u;                                           // tile_dim1=16 rows
    unsigned u5 = 256u;                                          // tensor_dim0_stride lo
    v4u g0; v8u g1;
    g0[0] = (unsigned)__builtin_amdgcn_readfirstlane((int)w0);
    g0[1] = (unsigned)__builtin_amdgcn_readfirstlane((int)w1);
    g0[2] = (unsigned)__builtin_amdgcn_readfirstlane((int)w2);
    g0[3] = (unsigned)__builtin_amdgcn_readfirstlane((int)w3);
    g1[0] = (unsigned)__builtin_amdgcn_readfirstlane((int)u0);
    g1[1] = (unsigned)__builtin_amdgcn_readfirstlane((int)u1);
    g1[2] = (unsigned)__builtin_amdgcn_readfirstlane((int)u2);
    g1[3] = (unsigned)__builtin_amdgcn_readfirstlane((int)u3);
    g1[4] = (unsigned)__builtin_amdgcn_readfirstlane((int)u4);
    g1[5] = (unsigned)__builtin_amdgcn_readfirstlane((int)u5);
    g1[6] = 0u;
    g1[7] = 0u;
    asm volatile("tensor_load_to_lds %0, %1" :: "s"(g0), "s"(g1) : "memory");
    __builtin_amdgcn_s_wait_tensorcnt(0);   // this wave's tile is in LDS
  }
  const float* tile = lds_tile + (unsigned)wave * (16u * ROWPITCH);
  for (int k = 0; k < 256; k += 4) {
    const int ka = k + half * 2;
    v2f a = *(const v2f*)(tile + l15 * ROWPITCH + ka);   // ds_load_b64, conflict-free
    v2f b = {};
    if (l15 == 0)      { b.x = W_att[ka]; b.y = W_att[ka + 1]; }
    else if (l15 == 1) { b.x = lin_W[ka]; b.y = lin_W[ka + 1]; }
    acc = __builtin_amdgcn_wmma_f32_16x16x4_f32(false, a, false, b,
                                                (short)0, acc, false, false);
  }
#else
  // host pass / non-TDM fallback: same math straight from global memory
  const long arow = rowBase + l15;
  for (int k = 0; k < 256; k += 4) {
    const int ka = k + half * 2;
    v2f a = *(const v2f*)(x + arow * 256 + ka);
    v2f b = {};
    if (l15 == 0)      { b.x = W_att[ka]; b.y = W_att[ka + 1]; }
    else if (l15 == 1) { b.x = lin_W[ka]; b.y = lin_W[ka + 1]; }
#if __has_builtin(__builtin_amdgcn_wmma_f32_16x16x4_f32)
    acc = __builtin_amdgcn_wmma_f32_16x16x4_f32(false, a, false, b,
                                                (short)0, acc, false, false);
#else
    acc[0] += a.x * b.x + a.y * b.y;   // host-pass placeholder only
#endif
  }
#endif
  // D layout: VGPR v, lanes 0-15: D[v][l15]; lanes 16-31: D[v+8][l15]
  if (l15 == 0) {
    const float s0 = att_src[0], d0 = att_dst[0];
    #pragma unroll
    for (int v = 0; v < 8; ++v) {
      long r = rowBase + v + 8 * half;
      float xl = acc[v];
      a_s[r] = xl * s0;
      a_d[r] = xl * d0;
    }
  } else if (l15 == 1) {
    const float b0 = lin_b[0];
    #pragma unroll
    for (int v = 0; v < 8; ++v) {
      long r = rowBase + v + 8 * half;
      x_t[r] = acc[v] + b0;
    }
  }
}

// ---------- node-array init
__global__ void jp_init_nodes(unsigned* m_u, float* s_sum, float* deg_d,
                              float* deg_s, int* mask, int n) {
  int i = blockIdx.x * blockDim.x + threadIdx.x;
  if (i < n) { m_u[i] = 0u; s_sum[i] = 0.f; deg_d[i] = 0.f; deg_s[i] = 0.f; mask[i] = -1; }
}
__global__ void jp_zero_f(float* y, int n) {
  int i = blockIdx.x * blockDim.x + threadIdx.x;
  if (i < n) y[i] = 0.f;
}

// combined edge list: i<E -> original edge (skip self-loops), i>=E -> self-loop (i-E)
__device__ __forceinline__ bool jp_edge(int i, int E, const int* src0, const int* dst0,
                                        int& s, int& d) {
  if (i < E) { s = src0[i]; d = dst0[i]; return s != d; }
  s = d = i - E;
  return true;
}

__global__ void jp_edge_max(const int* __restrict__ src0, const int* __restrict__ dst0,
                            int E, int n, const float* __restrict__ a_s,
                            const float* __restrict__ a_d, float* __restrict__ e_arr,
                            unsigned* __restrict__ m_u) {
  int EL = E + n;
  for (int i = blockIdx.x * blockDim.x + threadIdx.x; i < EL; i += gridDim.x * blockDim.x) {
    int s, d;
    if (!jp_edge(i, E, src0, dst0, s, d)) continue;
    float v = a_s[s] + a_d[d];
    float e = v > 0.f ? v : NEG_SLOPE * v;
    e_arr[i] = e;
    atomicMax(&m_u[d], fflip(e));
  }
}
__global__ void jp_edge_exp(const int* __restrict__ src0, const int* __restrict__ dst0,
                            int E, int n, float* __restrict__ e_arr,
                            const unsigned* __restrict__ m_u, float* __restrict__ s_sum) {
  int EL = E + n;
  for (int i = blockIdx.x * blockDim.x + threadIdx.x; i < EL; i += gridDim.x * blockDim.x) {
    int s, d;
    if (!jp_edge(i, E, src0, dst0, s, d)) continue;
    float ex = __expf(e_arr[i] - funflip(m_u[d]));
    e_arr[i] = ex;
    atomicAdd(&s_sum[d], ex);
  }
}
__global__ void jp_edge_alpha(const int* __restrict__ src0, const int* __restrict__ dst0,
                              int E, int n, float* __restrict__ e_arr,
                              const float* __restrict__ s_sum,
                              float* __restrict__ deg_d, float* __restrict__ deg_s) {
  int EL = E + n;
  for (int i = blockIdx.x * blockDim.x + threadIdx.x; i < EL; i += gridDim.x * blockDim.x) {
    int s, d;
    if (!jp_edge(i, E, src0, dst0, s, d)) continue;
    float alpha = e_arr[i] / (s_sum[d] + EPSF);
    e_arr[i] = alpha;
    atomicAdd(&deg_d[d], alpha);
    atomicAdd(&deg_s[s], alpha);
  }
}
__global__ void jp_edge_w(const int* __restrict__ src0, const int* __restrict__ dst0,
                          int E, int n, float* __restrict__ e_arr,
                          const float* __restrict__ deg_d, const float* __restrict__ deg_s) {
  int EL = E + n;
  for (int i = blockIdx.x * blockDim.x + threadIdx.x; i < EL; i += gridDim.x * blockDim.x) {
    int s, d;
    if (!jp_edge(i, E, src0, dst0, s, d)) continue;
    e_arr[i] *= rsqrtf(deg_d[d] + EPSF) * rsqrtf(deg_s[s] + EPSF);
  }
}

// y[dst] += w * xv[src]  (1-dim features -> scalar SpMV via atomics)
__global__ void jp_spmm(const int* __restrict__ src0, const int* __restrict__ dst0,
                        int E, int n, const float* __restrict__ w_arr,
                        const float* __restrict__ xv, float* __restrict__ y) {
  int EL = E + n;
  for (int i = blockIdx.x * blockDim.x + threadIdx.x; i < EL; i += gridDim.x * blockDim.x) {
    if (i + 8192 < E) {                 // speculative streaming prefetch of edge indices
      __builtin_prefetch(&src0[i + 8192], 0, 0);
      __builtin_prefetch(&dst0[i + 8192], 0, 0);
    }
    int s, d;
    if (!jp_edge(i, E, src0, dst0, s, d)) continue;
    atomicAdd(&y[d], w_arr[i] * xv[s]);
  }
}

// out = cy*y + ca*a + cb*b   (Jacobi recurrence combine)
__global__ void jp_recur(float* __restrict__ out, const float* __restrict__ y,
                         const float* __restrict__ a, const float* __restrict__ b,
                         float cy, float ca, float cb, int n) {
  int i = blockIdx.x * blockDim.x + threadIdx.x;
  if (i < n) out[i] = cy * y[i] + ca * a[i] + cb * b[i];
}

__global__ void jp_agg(const float* __restrict__ x_t, const float* __restrict__ xs1,
                       const float* __restrict__ xs2, const float* __restrict__ xs3,
                       const float* __restrict__ alphas_raw, float* __restrict__ agg, int n) {
  int i = blockIdx.x * blockDim.x + threadIdx.x;
  if (i >= n) return;
  float a0 = tanhf(alphas_raw[0]), a1 = tanhf(alphas_raw[1]);
  float a2 = tanhf(alphas_raw[2]), a3 = tanhf(alphas_raw[3]);
  agg[i] = (a0 + 1.0f) * x_t[i] + a1 * xs1[i] + a2 * xs2[i] + a3 * xs3[i];
}

// ---------- top-k via full bitonic sort on (~flip(value), idx) keys (ascending)
__global__ void jp_key(const float* __restrict__ agg, ull* __restrict__ key, int n, int pad) {
  int i = blockIdx.x * blockDim.x + threadIdx.x;
  if (i >= pad) return;
  key[i] = (i < n) ? ((((ull)(~fflip(agg[i]))) << 32) | (unsigned)i) : ~0ull;
}
__global__ void jp_bitonic(ull* __restrict__ key, int j, int kk, int pad) {
  int i = blockIdx.x * blockDim.x + threadIdx.x;
  int ixj = i ^ j;
  if (i < pad && ixj > i && ixj < pad) {
    ull a = key[i], b = key[ixj];
    bool up = ((i & kk) == 0);
    if ((a > b) == up) { key[i] = b; key[ixj] = a; }
  }
}
__global__ void jp_perm_mask(const ull* __restrict__ key, const float* __restrict__ agg,
                             int* __restrict__ perm, int* __restrict__ mask,
                             float* __restrict__ scale, int kp) {
  int i = blockIdx.x * blockDim.x + threadIdx.x;
  if (i < kp) {
    int p = (int)(unsigned)key[i];
    perm[i] = p;
    mask[p] = i;
    scale[i] = tanhf(agg[p]);
  }
}
__global__ void jp_gather(const float* __restrict__ x, const int* __restrict__ perm,
                          const float* __restrict__ scale, float* __restrict__ out, int kp) {
  long total = (long)kp * 256;
  long stride = (long)gridDim.x * blockDim.x;
  for (long t = (long)blockIdx.x * blockDim.x + threadIdx.x; t < total; t += stride) {
    int r = (int)(t >> 8), c = (int)(t & 255);
    out[t] = x[(long)perm[r] * 256 + c] * scale[r];
  }
}

// ---------- ordered edge compaction: count / scan / scatter
__global__ void jp_keep_count(const int* __restrict__ src0, const int* __restrict__ dst0,
                              const int* __restrict__ mask, int E, int* __restrict__ blk_cnt) {
  __shared__ int sh[TPB];
  int i = blockIdx.x * TPB + threadIdx.x;
  int keep = 0;
  if (i < E) keep = (mask[src0[i]] >= 0 && mask[dst0[i]] >= 0) ? 1 : 0;
  sh[threadIdx.x] = keep;
  __syncthreads();
  for (int off = TPB / 2; off > 0; off >>= 1) {
    if (threadIdx.x < off) sh[threadIdx.x] += sh[threadIdx.x + off];
    __syncthreads();
  }
  if (threadIdx.x == 0) blk_cnt[blockIdx.x] = sh[0];
}
__global__ void jp_scan_blocks(const int* __restrict__ blk_cnt, int* __restrict__ blk_off,
                               int* __restrict__ M_dev, int nblk) {
  if (blockIdx.x == 0 && threadIdx.x == 0) {
    int run = 0;
    for (int b = 0; b < nblk; ++b) { blk_off[b] = run; run += blk_cnt[b]; }
    *M_dev = run;
  }
}
__global__ void jp_scatter_edges(const int* __restrict__ src0, const int* __restrict__ dst0,
                                 const int* __restrict__ mask, const int* __restrict__ blk_off,
                                 const int* __restrict__ M_dev, int E,
                                 int* __restrict__ out_i, long base) {
  __shared__ int sh[TPB];
  int t = threadIdx.x;
  int i = blockIdx.x * TPB + t;
  int keep = 0, r = -1, c = -1;
  if (i < E) { r = mask[src0[i]]; c = mask[dst0[i]]; keep = (r >= 0 && c >= 0) ? 1 : 0; }
  sh[t] = keep;
  __syncthreads();
  for (int off = 1; off < TPB; off <<= 1) {     // inclusive Hillis-Steele
    int v = (t >= off) ? sh[t - off] : 0;
    __syncthreads();
    sh[t] += v;
    __syncthreads();
  }
  if (keep) {
    int pos = blk_off[blockIdx.x] + sh[t] - 1;
    long M = (long)(*M_dev);
    out_i[base + pos]     = r;   // edge_f row 0
    out_i[base + M + pos] = c;   // edge_f row 1
  }
}
__global__ void jp_finalize(const int* __restrict__ perm, const int* __restrict__ batch,
                            const int* __restrict__ M_dev, int kp,
                            int* __restrict__ out_i, long base) {
  int i = blockIdx.x * blockDim.x + threadIdx.x;
  if (i < kp) {
    long o = base + 2L * (long)(*M_dev);
    out_i[o + i]      = batch[perm[i]];  // batch_p
    out_i[o + kp + i] = perm[i];         // perm
  }
}

extern "C" void kernel_launch(void* const* d_in, const int* in_sizes, int n_in,
                              void* d_out, int out_size, void* d_ws, size_t ws_size,
                              hipStream_t stream) {
  (void)n_in; (void)out_size; (void)ws_size;
  const float* x          = (const float*)d_in[0];
  const int*   eidx       = (const int*)  d_in[1];
  const int*   batch      = (const int*)  d_in[2];
  const float* W_att      = (const float*)d_in[3];
  const float* att_src    = (const float*)d_in[4];
  const float* att_dst    = (const float*)d_in[5];
  const float* lin_W      = (const float*)d_in[6];
  const float* lin_b      = (const float*)d_in[7];
  const float* alphas_raw = (const float*)d_in[8];

  const int C = 256;
  const int N = in_sizes[0] / C;
  const int E = in_sizes[1] / 2;
  const int EL = E + N;
  const int kp = (int)ceil(0.8 * (double)N);
  int pad = 1; while (pad < N) pad <<= 1;
  const int* src0 = eidx;
  const int* dst0 = eidx + E;

  // workspace carve (256B-aligned)
  char* wp = (char*)d_ws;
  auto alloc = [&](size_t bytes) -> void* {
    void* p = (void*)wp; wp += (bytes + 255) & ~(size_t)255; return p;
  };
  float*    a_s    = (float*)   alloc((size_t)N * 4);
  float*    a_d    = (float*)   alloc((size_t)N * 4);
  float*    x_t    = (float*)   alloc((size_t)N * 4);
  unsigned* m_u    = (unsigned*)alloc((size_t)N * 4);
  float*    s_sum  = (float*)   alloc((size_t)N * 4);
  float*    deg_d  = (float*)   alloc((size_t)N * 4);
  float*    deg_s  = (float*)   alloc((size_t)N * 4);
  float*    e_arr  = (float*)   alloc((size_t)EL * 4);
  float*    y      = (float*)   alloc((size_t)N * 4);
  float*    xs1    = (float*)   alloc((size_t)N * 4);
  float*    xs2    = (float*)   alloc((size_t)N * 4);
  float*    xs3    = (float*)   alloc((size_t)N * 4);
  float*    agg    = (float*)   alloc((size_t)N * 4);
  ull*      key    = (ull*)     alloc((size_t)pad * 8);
  int*      perm   = (int*)     alloc((size_t)kp * 4);
  int*      mask   = (int*)     alloc((size_t)N * 4);
  float*    scale  = (float*)   alloc((size_t)kp * 4);
  int nEB = (E + TPB - 1) / TPB;
  int*      blkcnt = (int*)     alloc((size_t)nEB * 4);
  int*      blkoff = (int*)     alloc((size_t)nEB * 4);
  int*      M_dev  = (int*)     alloc(4);

  const int nbN = (N + TPB - 1) / TPB;
  const int nbE = (EL + TPB - 1) / TPB;

  // 1) init node accumulators + mask
  jp_init_nodes<<<nbN, TPB, 0, stream>>>(m_u, s_sum, deg_d, deg_s, mask, N);

  // 2) fused dual projection: TDM tile staging + WMMA (one wave per 16-row tile)
  {
    int waves = (N + 15) / 16;
    int blocks = (waves + 7) / 8;
    size_t ldsBytes = (size_t)8 * 16 * ROWPITCH * 4;   // 133,120 B dynamic LDS
    jp_dual_gemv_wmma<<<blocks, 256, ldsBytes, stream>>>(x, W_att, lin_W, att_src,
                                                         att_dst, lin_b, a_s, a_d,
                                                         x_t, N);
  }

  // 3) attention softmax + symmetric degree rescale (4 edge passes)
  jp_edge_max  <<<nbE, TPB, 0, stream>>>(src0, dst0, E, N, a_s, a_d, e_arr, m_u);
  jp_edge_exp  <<<nbE, TPB, 0, stream>>>(src0, dst0, E, N, e_arr, m_u, s_sum);
  jp_edge_alpha<<<nbE, TPB, 0, stream>>>(src0, dst0, E, N, e_arr, s_sum, deg_d, deg_s);
  jp_edge_w    <<<nbE, TPB, 0, stream>>>(src0, dst0, E, N, e_arr, deg_d, deg_s);

  // 4) Jacobi recurrence (AJ=BJ=1, L=-1, R=1, K=3)
  const double AJd = 1.0, BJd = 1.0, Ld = -1.0, Rd = 1.0;
  double coef1 = (AJd - BJd) / 2.0 - (AJd + BJd + 2.0) / 2.0 * (Ld + Rd) / (Rd - Ld);
  double coef2 = (AJd + BJd + 2.0) / (Rd - Ld);
  float* xs[4] = { x_t, xs1, xs2, xs3 };

  jp_zero_f<<<nbN, TPB, 0, stream>>>(y, N);
  jp_spmm  <<<nbE, TPB, 0, stream>>>(src0, dst0, E, N, e_arr, xs[0], y);
  jp_recur <<<nbN, TPB, 0, stream>>>(xs[1], y, xs[0], xs[0],
                                     (float)coef2, (float)coef1, 0.f, N);
  for (int k = 2; k <= 3; ++k) {
    double coef_l    = 2.0 * k * (k + AJd + BJd) * (2.0 * k - 2.0 + AJd + BJd);
    double coef_lm11 = (2.0 * k + AJd + BJd - 1.0) * (2.0 * k + AJd + BJd) * (2.0 * k + AJd + BJd - 2.0);
    double coef_lm12 = (2.0 * k + AJd + BJd - 1.0) * (AJd * AJd - BJd * BJd);
    double coef_lm2  = 2.0 * (k - 1.0 + AJd) * (k - 1.0 + BJd) * (2.0 * k + AJd + BJd);
    double tmp1 = coef_lm11 / coef_l, tmp2 = coef_lm12 / coef_l, tmp3 = coef_lm2 / coef_l;
    double tmp1_2 = tmp1 * (2.0 / (Rd - Ld));
    double tmp2_2 = tmp1 * ((Rd + Ld) / (Rd - Ld)) + tmp2;
    jp_zero_f<<<nbN, TPB, 0, stream>>>(y, N);
    jp_spmm  <<<nbE, TPB, 0, stream>>>(src0, dst0, E, N, e_arr, xs[k - 1], y);
    jp_recur <<<nbN, TPB, 0, stream>>>(xs[k], y, xs[k - 1], xs[k - 2],
                                       (float)tmp1_2, (float)(-tmp2_2), (float)(-tmp3), N);
  }

  // 5) aggregate score
  jp_agg<<<nbN, TPB, 0, stream>>>(x_t, xs1, xs2, xs3, alphas_raw, agg, N);

  // 6) descending top-k via bitonic sort on 64-bit keys
  int nbP = pad / TPB;
  jp_key<<<nbP, TPB, 0, stream>>>(agg, key, N, pad);
  for (int kk = 2; kk <= pad; kk <<= 1)
    for (int j = kk >> 1; j > 0; j >>= 1)
      jp_bitonic<<<nbP, TPB, 0, stream>>>(key, j, kk, pad);

  // 7) perm / mask / gather x_p into d_out[0 .. kp*256)
  int nbK = (kp + TPB - 1) / TPB;
  jp_perm_mask<<<nbK, TPB, 0, stream>>>(key, agg, perm, mask, scale, kp);
  {
    long total = (long)kp * 256;
    int blocks = (int)((total + TPB - 1) / TPB);
    if (blocks > 65535 * 16) blocks = 65535 * 16;
    jp_gather<<<blocks, TPB, 0, stream>>>(x, perm, scale, (float*)d_out, kp);
  }

  // 8) ordered edge compaction into d_out (ints bit-stored after x_p)
  int* out_i = (int*)d_out;
  long base = (long)kp * 256;
  jp_keep_count  <<<nEB, TPB, 0, stream>>>(src0, dst0, mask, E, blkcnt);
  jp_scan_blocks <<<1, 32, 0, stream>>>(blkcnt, blkoff, M_dev, nEB);
  jp_scatter_edges<<<nEB, TPB, 0, stream>>>(src0, dst0, mask, blkoff, M_dev, E, out_i, base);
  jp_finalize    <<<nbK, TPB, 0, stream>>>(perm, batch, M_dev, kp, out_i, base);
}